// _RoIPooling_67001489817768
// MI455X (gfx1250) — compile-verified
//
#include <hip/hip_runtime.h>
#include <math.h>

// ---------------- RoIPool (Fast R-CNN max pool) for MI455X / gfx1250 ----------------
// Strategy: one block per (roi, 8-channel group). The ROI's clamped bounding box
// (nrows x ncols, row stride W) is DMA'd global->LDS with the CDNA5 Tensor Data
// Mover (tensor_load_to_lds), double-buffered so channel c+1's copy overlaps the
// max-reduction of channel c. Lanes 0..48 each own one 7x7 output bin.

typedef unsigned int u32x4 __attribute__((ext_vector_type(4)));
typedef int          i32x8 __attribute__((ext_vector_type(8)));
typedef int          i32x4 __attribute__((ext_vector_type(4)));

#define PH 7
#define PW 7
static constexpr int   C_DIM = 256;
static constexpr int   H_DIM = 64;
static constexpr int   W_DIM = 64;
static constexpr float SCALE = 0.0625f;
static constexpr int   CPB   = 8;                 // channels per block
static constexpr int   TILE_ELEMS = H_DIM * W_DIM; // worst-case bounding box

// LDS byte offset of a __shared__ pointer: flat addr low 32 bits are the LDS
// offset (ISA 10.2).
__device__ __forceinline__ unsigned lds_off(const void* p) {
    return (unsigned)(unsigned long long)(p);
}

// Issue one 2-D TDM tile load: copies nrows rows of ncols floats from global memory
//   at gaddr (row stride stride_elems floats) to LDS at lds_byte, rows packed ncols floats apart.
// D# bit layout per CDNA5 ISA sec 8.3/8.4. Group2/3 zero (<=2D tensor).
__device__ __forceinline__ void tdm_load_2d(unsigned lds_byte,
                                            unsigned long long gaddr,
                                            int ncols, int nrows, int stride_elems) {
    unsigned glo = (unsigned)(gaddr & 0xFFFFFFFFull);
    unsigned ghi = (unsigned)((gaddr >> 32) & 0x1FFFFFFull);
    u32x4 g0;
    g0[0] = (unsigned)__builtin_amdgcn_readfirstlane((int)1u);           // count=1, user desc
    g0[1] = (unsigned)__builtin_amdgcn_readfirstlane((int)lds_byte);     // lds_addr (bytes)
    g0[2] = (unsigned)__builtin_amdgcn_readfirstlane((int)glo);          // global_addr[31:0]
    g0[3] = (unsigned)__builtin_amdgcn_readfirstlane((int)(ghi | (2u << 30))); // ga[56:32] | type=2
    i32x8 g1;
    g1[0] = __builtin_amdgcn_readfirstlane(2 << 16);                     // data_size=4B, mask=0
    g1[1] = __builtin_amdgcn_readfirstlane((ncols & 0xFFFF) << 16);      // tensor_dim0[15:0]
    g1[2] = __builtin_amdgcn_readfirstlane((int)(((unsigned)ncols >> 16) |
                                                 ((unsigned)(nrows & 0xFFFF) << 16))); // td0 hi | td1 lo
    g1[3] = __builtin_amdgcn_readfirstlane((int)(((unsigned)nrows >> 16) |
                                                 ((unsigned)(ncols & 0xFFFF) << 16))); // td1 hi | tile_dim0
    g1[4] = __builtin_amdgcn_readfirstlane(nrows & 0xFFFF);              // tile_dim1 (tile_dim2=0)
    g1[5] = __builtin_amdgcn_readfirstlane(stride_elems);                // tensor_dim0_stride[31:0]
    g1[6] = 0;                                                           // stride hi | dim1_stride lo
    g1[7] = 0;
    i32x4 gz4; gz4[0] = 0; gz4[1] = 0; gz4[2] = 0; gz4[3] = 0;
    i32x8 gz8; gz8[0] = 0; gz8[1] = 0; gz8[2] = 0; gz8[3] = 0;
    gz8[4] = 0; gz8[5] = 0; gz8[6] = 0; gz8[7] = 0;
    // amdgpu-toolchain (clang-23) 6-arg form:
    // (uint32x4 g0, int32x8 g1, int32x4, int32x4, int32x8, i32 cpol)
    __builtin_amdgcn_tensor_load_to_lds(g0, g1, gz4, gz4, gz8, 0);
}

__global__ void __launch_bounds__(64)
roipool_tdm_kernel(const float* __restrict__ feat,
                   const float* __restrict__ rois,
                   float* __restrict__ out) {
    __shared__ float tiles[2][TILE_ELEMS];   // 32 KB double buffer

    const int r   = blockIdx.x;
    const int c0  = blockIdx.y * CPB;
    const int tid = threadIdx.x;

    // ROI decode (matches jnp: round = half-to-even via rintf, int cast after)
    const float* roi = rois + (size_t)r * 5;
    const int bidx = (int)roi[0];
    const int sw = (int)rintf(roi[1] * SCALE);
    const int sh = (int)rintf(roi[2] * SCALE);
    const int ew = (int)rintf(roi[3] * SCALE);
    const int eh = (int)rintf(roi[4] * SCALE);
    const float roi_w = (float)max(ew - sw + 1, 1);
    const float roi_h = (float)max(eh - sh + 1, 1);
    const float bh = roi_h / (float)PH;
    const float bw = roi_w / (float)PW;

    // Clamped bounding box staged into LDS
    const int hlo = min(max(sh, 0), H_DIM);
    const int hhi = min(max(eh + 1, 0), H_DIM);
    const int wlo = min(max(sw, 0), W_DIM);
    const int whi = min(max(ew + 1, 0), W_DIM);
    const int nrows = hhi - hlo, ncols = whi - wlo;
    const bool region = (nrows > 0) && (ncols > 0);

    // Per-bin bounds (identical across the 8 channels -> hoist out of loop)
    const int ph = tid / PW, pw = tid % PW;
    int hs = 0, he = 0, ws = 0, we = 0;
    bool empty = true;
    if (tid < PH * PW) {
        hs = min(max((int)floorf((float)ph * bh) + sh, 0), H_DIM);
        he = min(max((int)ceilf(((float)ph + 1.0f) * bh) + sh, 0), H_DIM);
        ws = min(max((int)floorf((float)pw * bw) + sw, 0), W_DIM);
        we = min(max((int)ceilf(((float)pw + 1.0f) * bw) + sw, 0), W_DIM);
        empty = (he <= hs) || (we <= ws);
    }

    const float* base = feat + ((size_t)bidx * C_DIM + c0) * (H_DIM * W_DIM)
                             + (size_t)hlo * W_DIM + wlo;
    const bool loader = (tid < 32);   // wave 0 drives the TDM pipeline

    if (region && loader) {           // prologue: stage channel c0 into buffer 0
        tdm_load_2d(lds_off(&tiles[0][0]), (unsigned long long)base,
                    ncols, nrows, W_DIM);
    }

    for (int i = 0; i < CPB; ++i) {
        if (region && loader) {
            if (i + 1 < CPB) {        // stage next channel, wait for current
                tdm_load_2d(lds_off(&tiles[(i + 1) & 1][0]),
                            (unsigned long long)(base + (size_t)(i + 1) * H_DIM * W_DIM),
                            ncols, nrows, W_DIM);
                __builtin_amdgcn_s_wait_tensorcnt(1);
            } else {
                __builtin_amdgcn_s_wait_tensorcnt(0);
            }
        }
        __syncthreads();              // publish LDS tile to wave 1

        if (tid < PH * PW) {
            float m = 0.0f;           // empty bins -> 0 (reference semantics)
            if (!empty) {
                const float* t = &tiles[i & 1][0];
                m = -__builtin_inff();
                for (int h = hs; h < he; ++h) {
                    const int rowoff = (h - hlo) * ncols - wlo;
                    for (int w = ws; w < we; ++w)
                        m = fmaxf(m, t[rowoff + w]);
                }
            }
            out[((size_t)r * C_DIM + (c0 + i)) * (PH * PW) + tid] = m;
        }
        __syncthreads();              // protect buffer (i&1) before it is re-filled
    }
}

extern "C" void kernel_launch(void* const* d_in, const int* in_sizes, int n_in,
                              void* d_out, int out_size, void* d_ws, size_t ws_size,
                              hipStream_t stream) {
    const float* feat = (const float*)d_in[0];
    const float* rois = (const float*)d_in[1];
    float* out = (float*)d_out;
    const int R = in_sizes[1] / 5;                 // (R,5) roi rows
    dim3 grid((unsigned)R, (unsigned)(C_DIM / CPB));
    roipool_tdm_kernel<<<grid, 64, 0, stream>>>(feat, rois, out);
}